// CombinedLoss_79645873537166
// MI455X (gfx1250) — compile-verified
//
#include <hip/hip_runtime.h>
#include <hip/hip_bf16.h>
#include <math.h>

typedef float v2f __attribute__((ext_vector_type(2)));
typedef float v8f __attribute__((ext_vector_type(8)));

#define Bdim 8
#define Hd 1024
#define Wd 1024
#define NTOT (Bdim * Hd * Wd)   // 8388608
#define SMOOTH_NR 1e-5f

// red[] layout (floats), at d_ws + 2*NTOT:
//   [0,4096)      min(m) per blur_x block
//   [4096,8192)   max(m) per blur_x block
//   [8192,9216)   max(sm) per blur_b block (1024 blocks)
//   [9216..9218]  finals: maxsm, minm, maxm
//   [9472, +7*8192) loss partials (red2)
#define RED_FIN 9216
#define RED2_OFF 9472

// ---------------- Gaussian weights (sigma=2, radius=8, 17 taps) ----------------
__device__ __forceinline__ float gauss_inv_sum() {
    float s = 0.f;
#pragma unroll
    for (int t = 0; t < 17; ++t) {
        float d = (float)(t - 8);
        s += __expf(-0.125f * d * d);
    }
    return 1.0f / s;
}
__device__ __forceinline__ float band_w(int d, float inv) {
    float dd = (float)(d - 8);
    return (d >= 0 && d <= 16) ? __expf(-0.125f * dd * dd) * inv : 0.0f;
}

// ---------------- CDNA5 async global->LDS copy (512B/wave/instr) ---------------
__device__ __forceinline__ void async_ld_b128(unsigned lds_off, const float* g) {
    asm volatile("global_load_async_to_lds_b128 %0, %1, off"
                 :: "v"(lds_off), "v"(g) : "memory");
}
__device__ __forceinline__ void wait_async0() {
    asm volatile("s_wait_asynccnt 0x0" ::: "memory");
}

// ---------------- WMMA wrapper: D = A(16x4) * B(4x16) + C, f32 -----------------
__device__ __forceinline__ v8f wmma4(v2f a, v2f b, v8f c) {
    return __builtin_amdgcn_wmma_f32_16x16x4_f32(false, a, false, b,
                                                 (short)0, c, false, false);
}

// ---------------- Block reduction helpers (blockDim == 256) --------------------
__device__ __forceinline__ float blkSum(float v, float* s) {
    int t = threadIdx.x; s[t] = v; __syncthreads();
    for (int k = 128; k > 0; k >>= 1) { if (t < k) s[t] += s[t + k]; __syncthreads(); }
    float r = s[0]; __syncthreads(); return r;
}
__device__ __forceinline__ float blkMax(float v, float* s) {
    int t = threadIdx.x; s[t] = v; __syncthreads();
    for (int k = 128; k > 0; k >>= 1) { if (t < k) s[t] = fmaxf(s[t], s[t + k]); __syncthreads(); }
    float r = s[0]; __syncthreads(); return r;
}
__device__ __forceinline__ float blkMin(float v, float* s) {
    int t = threadIdx.x; s[t] = v; __syncthreads();
    for (int k = 128; k > 0; k >>= 1) { if (t < k) s[t] = fminf(s[t], s[t + k]); __syncthreads(); }
    float r = s[0]; __syncthreads(); return r;
}

// ---------------- Pass 1: blur along X via WMMA + fold min/max of (t>0) --------
// Block: 16 rows x 128 cols (8 waves x one 16x16 tile). Grid (8, 64, 8).
__global__ void k_blur_x(const float* __restrict__ in, float* __restrict__ out,
                         float* __restrict__ red) {
    __shared__ float lds[16 * 144];  // 16 x (128 + 2*8), reflect-padded
    __shared__ float sdata[256];
    const int tid  = threadIdx.x;
    const int b    = blockIdx.z;
    const int row0 = blockIdx.y * 16;
    const int x0   = blockIdx.x * 128;

    if (x0 >= 8 && x0 + 136 <= Wd) {
        // interior: contiguous, 16B-aligned -> async B128 straight to LDS
        for (int e = tid; e < 16 * 36; e += 256) {
            int r = e / 36, cc = (e - r * 36) * 4;
            const float* g = &in[((size_t)b * Hd + row0 + r) * Wd + (x0 - 8 + cc)];
            async_ld_b128((unsigned)(size_t)&lds[r * 144 + cc], g);
        }
        wait_async0();
    } else {
        // edge blocks: scalar with 'symmetric' reflection
        for (int e = tid; e < 16 * 144; e += 256) {
            int r = e / 144, c = e - r * 144;
            int g = x0 - 8 + c;
            g = (g < 0) ? (-1 - g) : ((g >= Wd) ? (2 * Wd - 1 - g) : g);
            lds[e] = in[((size_t)b * Hd + row0 + r) * Wd + g];
        }
    }
    __syncthreads();

    // fold global min/max of (t>0) over this tile (duplicates are harmless)
    {
        float mn = 3.4e38f, mx = -3.4e38f;
        for (int e = tid; e < 16 * 144; e += 256) {
            float m = (lds[e] > 0.f) ? 1.f : 0.f;
            mn = fminf(mn, m); mx = fmaxf(mx, m);
        }
        mn = blkMin(mn, sdata);
        mx = blkMax(mx, sdata);
        if (tid == 0) {
            int gb = (blockIdx.z * 64 + blockIdx.y) * 8 + blockIdx.x;  // 0..4095
            red[gb] = mn; red[4096 + gb] = mx;
        }
    }

    const float inv = gauss_inv_sum();
    const int lane = tid & 31, w = tid >> 5;
    const int m = lane & 15, half = lane >> 4;

    v2f bf[8];  // Kband[i][n] = w[i-n], i = 4g + v + 2*half, n = m
#pragma unroll
    for (int g = 0; g < 8; ++g)
#pragma unroll
        for (int v = 0; v < 2; ++v) {
            int i = 4 * g + v + 2 * half;
            bf[g][v] = band_w(i - m, inv);
        }

    v8f c = {};
#pragma unroll
    for (int g = 0; g < 8; ++g) {
        v2f a;
        a[0] = lds[m * 144 + 16 * w + 4 * g + 2 * half + 0];
        a[1] = lds[m * 144 + 16 * w + 4 * g + 2 * half + 1];
        c = wmma4(a, bf[g], c);
    }

#pragma unroll
    for (int r = 0; r < 8; ++r) {
        int row = row0 + r + 8 * half;
        out[((size_t)b * Hd + row) * Wd + x0 + 16 * w + m] = c[r];
    }
}

// ---------------- Pass 2: blur along Y via WMMA, async-LDS for all blocks ------
__global__ void k_blur_y(const float* __restrict__ in, float* __restrict__ out) {
    __shared__ float lds[32 * 128];
    const int tid = threadIdx.x;
    const int b   = blockIdx.z;
    const int y0  = blockIdx.y * 16;
    const int x0  = blockIdx.x * 128;

    // y-reflection only remaps the row; each row stays contiguous & aligned.
    for (int e = tid; e < 32 * 32; e += 256) {
        int r = e >> 5, cc = (e & 31) * 4;
        int gy = y0 - 8 + r;
        gy = (gy < 0) ? (-1 - gy) : ((gy >= Hd) ? (2 * Hd - 1 - gy) : gy);
        const float* g = &in[((size_t)b * Hd + gy) * Wd + x0 + cc];
        async_ld_b128((unsigned)(size_t)&lds[r * 128 + cc], g);
    }
    wait_async0();
    __syncthreads();

    const float inv = gauss_inv_sum();
    const int lane = tid & 31, w = tid >> 5;
    const int m = lane & 15, half = lane >> 4;

    v2f af[8];  // Kband[m][k] = w[k-m], k = 4g + v + 2*half
#pragma unroll
    for (int g = 0; g < 8; ++g)
#pragma unroll
        for (int v = 0; v < 2; ++v) {
            int k = 4 * g + v + 2 * half;
            af[g][v] = band_w(k - m, inv);
        }

    v8f c = {};
#pragma unroll
    for (int g = 0; g < 8; ++g) {
        v2f bfr;
        bfr[0] = lds[(4 * g + 2 * half + 0) * 128 + 16 * w + m];
        bfr[1] = lds[(4 * g + 2 * half + 1) * 128 + 16 * w + m];
        c = wmma4(af[g], bfr, c);
    }

#pragma unroll
    for (int r = 0; r < 8; ++r) {
        int row = y0 + r + 8 * half;
        out[((size_t)b * Hd + row) * Wd + x0 + 16 * w + m] = c[r];
    }
}

// ---------------- Pass 3: batch-axis blur (L=8, reflect r=8) + fold max(sm) ----
// float4 per thread; grid 1024 blocks x 256 threads covers the 1M plane.
__global__ void k_blur_b(const float* __restrict__ in, float* __restrict__ out,
                         float* __restrict__ red) {
    __shared__ float sdata[256];
    const int tid = threadIdx.x;
    const int j = (blockIdx.x * 256 + tid) * 4;
    const float inv = gauss_inv_sum();
    float wq[17];
#pragma unroll
    for (int t = 0; t < 17; ++t) {
        float d = (float)(t - 8);
        wq[t] = __expf(-0.125f * d * d) * inv;
    }
    float4 v[8];
#pragma unroll
    for (int bb = 0; bb < 8; ++bb)
        v[bb] = *(const float4*)&in[((size_t)bb << 20) + j];

    float mxs = -3.4e38f;
#pragma unroll
    for (int b = 0; b < 8; ++b) {
        float4 s = {0.f, 0.f, 0.f, 0.f};
#pragma unroll
        for (int t = 0; t < 17; ++t) {
            int q = b + t - 8;
            q = (q < 0) ? (-1 - q) : ((q > 7) ? (15 - q) : q);
            s.x += wq[t] * v[q].x; s.y += wq[t] * v[q].y;
            s.z += wq[t] * v[q].z; s.w += wq[t] * v[q].w;
        }
        *(float4*)&out[((size_t)b << 20) + j] = s;
        mxs = fmaxf(mxs, fmaxf(fmaxf(s.x, s.y), fmaxf(s.z, s.w)));
    }
    mxs = blkMax(mxs, sdata);
    if (tid == 0) red[8192 + blockIdx.x] = mxs;
}

// ---------------- Pass 4: finalize the three global scalars --------------------
__global__ void k_minmax_fin(float* __restrict__ red) {
    __shared__ float sdata[256];
    const int tid = threadIdx.x;
    float mn = 3.4e38f, mx = -3.4e38f, ms = -3.4e38f;
    for (int jj = tid; jj < 4096; jj += 256) {
        mn = fminf(mn, red[jj]);
        mx = fmaxf(mx, red[4096 + jj]);
    }
    for (int jj = tid; jj < 1024; jj += 256) ms = fmaxf(ms, red[8192 + jj]);
    ms = blkMax(ms, sdata); mn = blkMin(mn, sdata); mx = blkMax(mx, sdata);
    if (tid == 0) { red[RED_FIN] = ms; red[RED_FIN + 1] = mn; red[RED_FIN + 2] = mx; }
}

// ---------------- Pass 5: fused elementwise losses (float4), partials ----------
// Grid (1024, 8): blockIdx.y = batch; block covers 1024 contiguous elements.
__global__ void k_main(const float* __restrict__ pred, const float* __restrict__ target,
                       const float* __restrict__ second, const float* __restrict__ sm,
                       const float* __restrict__ scal, float* __restrict__ red2) {
    __shared__ float sdata[256];
    const int tid = threadIdx.x;
    const int b = blockIdx.y;
    const float maxsm = scal[0], minm = scal[1], maxm = scal[2];
    const float inv_mx = 1.0f / (maxsm + 1e-8f);
    const int   smz    = (maxsm < 1e-8f);
    const float inv_mm = 1.0f / (maxm - minm + 1e-8f);
    const size_t i = ((size_t)b << 20) + (size_t)blockIdx.x * 1024 + (size_t)tid * 4;

    float4 x4 = *(const float4*)&pred[i];
    float4 t4 = *(const float4*)&target[i];
    float4 s4 = *(const float4*)&second[i];
    float4 m4 = *(const float4*)&sm[i];
    float xs[4] = {x4.x, x4.y, x4.z, x4.w};
    float tsv[4] = {t4.x, t4.y, t4.z, t4.w};
    float sp[4] = {s4.x, s4.y, s4.z, s4.w};
    float sv[4] = {m4.x, m4.y, m4.z, m4.w};

    float inter = 0.f, psq = 0.f, tsq = 0.f, bce = 0.f, bcs = 0.f, kls = 0.f, dv = 0.f;
#pragma unroll
    for (int c = 0; c < 4; ++c) {
        float x = xs[c], t = tsv[c], spv = sp[c], smv = sv[c];
        float p  = 1.0f / (1.0f + __expf(-x));
        float sx = log1pf(__expf(-fabsf(x)));
        bce += fmaxf(x, 0.f) - x * t + sx;
        float m   = (t > 0.f) ? 1.f : 0.f;
        float tsi = (m - minm) * inv_mm;
        bcs += fmaxf(x, 0.f) - x * tsi + sx;
        float ts = smz ? 0.f : smv * inv_mx;
        float a0 = 1.f - p, a1 = p;
        float lse = fmaxf(a0, a1) + log1pf(__expf(-fabsf(a0 - a1)));
        float t0 = 1.f - ts, t1 = ts;
        if (t0 > 0.f) kls += t0 * (__logf(t0) - (a0 - lse));
        if (t1 > 0.f) kls += t1 * (__logf(t1) - (a1 - lse));
        float d = x - spv; dv += d * d;
        inter += p * t; psq += p * p; tsq += t * t;
    }
    const int gblk = b * 1024 + blockIdx.x;
    float r;
    r = blkSum(inter, sdata); if (tid == 0) red2[0 * 8192 + gblk] = r;
    r = blkSum(psq,   sdata); if (tid == 0) red2[1 * 8192 + gblk] = r;
    r = blkSum(tsq,   sdata); if (tid == 0) red2[2 * 8192 + gblk] = r;
    r = blkSum(bce,   sdata); if (tid == 0) red2[3 * 8192 + gblk] = r;
    r = blkSum(bcs,   sdata); if (tid == 0) red2[4 * 8192 + gblk] = r;
    r = blkSum(kls,   sdata); if (tid == 0) red2[5 * 8192 + gblk] = r;
    r = blkSum(dv,    sdata); if (tid == 0) red2[6 * 8192 + gblk] = r;
}

// ---------------- Pass 6: final deterministic combine --------------------------
__global__ void k_final(const float* __restrict__ red2, const float* __restrict__ red,
                        float* __restrict__ out) {
    __shared__ float sdata[256];
    const int tid = threadIdx.x;
    float dice_sum = 0.f;
    for (int b = 0; b < 8; ++b) {
        float i_ = 0.f, p_ = 0.f, t_ = 0.f;
        for (int j = tid; j < 1024; j += 256) {
            int g = b * 1024 + j;
            i_ += red2[0 * 8192 + g];
            p_ += red2[1 * 8192 + g];
            t_ += red2[2 * 8192 + g];
        }
        float inter = blkSum(i_, sdata);
        float psq   = blkSum(p_, sdata);
        float tsq   = blkSum(t_, sdata);
        dice_sum += 1.0f - (2.0f * inter + SMOOTH_NR) / (psq + tsq + SMOOTH_NR);
    }
    float dice = dice_sum * (1.0f / 8.0f);

    float sums[4];
    for (int k = 0; k < 4; ++k) {
        float s = 0.f;
        for (int j = tid; j < 8192; j += 256) s += red2[(3 + k) * 8192 + j];
        sums[k] = blkSum(s, sdata);
    }
    const float invN = 1.0f / (float)NTOT;
    float bce = sums[0] * invN, bcesoft = sums[1] * invN;
    float kl  = sums[2] * invN, div = sums[3] * invN;
    kl = fminf(fmaxf(kl, 0.f), 2.f);
    if (red[RED_FIN] < 1e-8f) kl = 0.f;   // all(ts == 0) <=> max(sm) < 1e-8
    if (tid == 0)
        out[0] = (dice + bce + 0.5f * kl + 0.5f * div + 0.5f * bcesoft) * (1.0f / 3.5f);
}

// ---------------- Host side ----------------------------------------------------
extern "C" void kernel_launch(void* const* d_in, const int* in_sizes, int n_in,
                              void* d_out, int out_size, void* d_ws, size_t ws_size,
                              hipStream_t stream) {
    // setup_inputs order: image (unused), pred, target, second_pred
    const float* pred   = (const float*)d_in[1];
    const float* target = (const float*)d_in[2];
    const float* second = (const float*)d_in[3];
    float* out = (float*)d_out;

    float* wsA  = (float*)d_ws;                    // N floats
    float* wsB  = wsA + (size_t)NTOT;              // N floats
    float* red  = wsB + (size_t)NTOT;              // partials + finals
    float* red2 = red + RED2_OFF;                  // 7 * 8192 loss partials

    dim3 blurGrid(Wd / 128, Hd / 16, Bdim);        // (8, 64, 8)

    k_blur_x<<<blurGrid, 256, 0, stream>>>(target, wsA, red);
    k_blur_y<<<blurGrid, 256, 0, stream>>>(wsA, wsB);
    k_blur_b<<<dim3(1024), 256, 0, stream>>>(wsB, wsA, red);   // sm -> wsA
    k_minmax_fin<<<dim3(1), 256, 0, stream>>>(red);

    k_main<<<dim3(1024, Bdim), 256, 0, stream>>>(pred, target, second, wsA,
                                                 red + RED_FIN, red2);
    k_final<<<dim3(1), 256, 0, stream>>>(red2, red, out);
}